// IMGSR_27350351741231
// MI455X (gfx1250) — compile-verified
//
#include <hip/hip_runtime.h>

// Problem dims (fixed by reference setup_inputs()).
#define B_   1
#define C_   16
#define V_   360
#define D_   736
#define P_   16384
#define VD_  (V_ * D_)
#define BLOCK 256
#define VSPLIT 8

#define AS1 __attribute__((address_space(1)))
#define AS3 __attribute__((address_space(3)))

typedef int v2i_ __attribute__((vector_size(2 * sizeof(int))));

#if defined(__HIP_DEVICE_COMPILE__) && \
    __has_builtin(__builtin_amdgcn_global_load_async_to_lds_b64) && \
    __has_builtin(__builtin_amdgcn_global_load_async_to_lds_b32)
#define HAVE_ASYNC 1
#else
#define HAVE_ASYNC 0
#endif

template <int N>
__device__ __forceinline__ void wait_async() {
#if defined(__HIP_DEVICE_COMPILE__) && __has_builtin(__builtin_amdgcn_s_wait_asynccnt)
    __builtin_amdgcn_s_wait_asynccnt(N);
#elif defined(__HIP_DEVICE_COMPILE__)
    asm volatile("s_wait_asynccnt %0" ::"i"(N) : "memory");
#endif
}

// Back-projection: each thread owns one recon pixel p, loops over a view
// segment, bilinear-gathers 2x2 x 16ch from the sinogram, accumulates.
// VPER = views per segment (compile-time: no runtime division).
template <int VPER>
__global__ __launch_bounds__(BLOCK) void IMGSR_bp_kernel(
    const float* __restrict__ sino,   // [C_,V_,D_]
    const float* __restrict__ grid,   // [V_,P_,2]
    const float* __restrict__ sq,     // [V_,P_]
    float* __restrict__ dstbase)      // [nseg][C_,P_]
{
    const int p   = blockIdx.x * BLOCK + threadIdx.x;
    const int seg = blockIdx.y;
    const int v0  = seg * VPER;
    const int v1  = v0 + VPER;

    const float2* __restrict__ g2 = reinterpret_cast<const float2*>(grid);

    float acc[C_];
#pragma unroll
    for (int c = 0; c < C_; ++c) acc[c] = 0.0f;

#if HAVE_ASYNC
    // Per-wave async staging of the streamed operands, prefetch distance 2
    // (4 rotating buffers). Each lane stages & consumes only its own element
    // -> per-wave ASYNCcnt is the only sync needed, no workgroup barriers.
    __shared__ float2 gbuf[4][BLOCK];
    __shared__ float  qbuf[4][BLOCK];

    auto stage = [&](int v) {
        const int b = v & 3;
        __builtin_amdgcn_global_load_async_to_lds_b64(
            (AS1 v2i_*)(g2 + (size_t)v * P_ + p),
            (AS3 v2i_*)(&gbuf[b][threadIdx.x]), 0, 0);
        __builtin_amdgcn_global_load_async_to_lds_b32(
            (AS1 int*)(sq + (size_t)v * P_ + p),
            (AS3 int*)(&qbuf[b][threadIdx.x]), 0, 0);
    };
    stage(v0);
    if (VPER > 1) stage(v0 + 1);
#endif

    for (int v = v0; v < v1; ++v) {
        float2 g;
        float  q;
#if HAVE_ASYNC
        // In-order async completion: waiting ASYNCcnt<=2*pending_ahead
        // guarantees view v's two ops have landed in LDS.
        if (v + 2 < v1) {
            stage(v + 2);
            wait_async<4>();
        } else if (v + 1 < v1) {
            wait_async<2>();
        } else {
            wait_async<0>();
        }
        g = gbuf[v & 3][threadIdx.x];
        q = qbuf[v & 3][threadIdx.x];
#else
        if (v + 8 < v1) {
            __builtin_prefetch(g2 + (size_t)(v + 8) * P_ + p, 0, 1);
            __builtin_prefetch(sq + (size_t)(v + 8) * P_ + p, 0, 1);
        }
        g = g2[(size_t)v * P_ + p];
        q = sq[(size_t)v * P_ + p];
#endif
        // Unnormalize (align_corners=False): ix = ((gx+1)*W - 1) * 0.5
        const float ix = (g.x + 1.0f) * (0.5f * D_) - 0.5f;
        const float iy = (g.y + 1.0f) * (0.5f * V_) - 0.5f;
        const float fx = floorf(ix), fy = floorf(iy);
        const float wx = ix - fx,    wy = iy - fy;
        const int ifx = (int)fx, ify = (int)fy;
        const int x0 = min(max(ifx,     0), D_ - 1);
        const int x1 = min(max(ifx + 1, 0), D_ - 1);
        const int y0 = min(max(ify,     0), V_ - 1);
        const int y1 = min(max(ify + 1, 0), V_ - 1);

        const float s   = q * 1000.0f;
        const float omx = 1.0f - wx, omy = 1.0f - wy;
        const float w00 = omx * omy * s;
        const float w01 = wx  * omy * s;
        const float w10 = omx * wy  * s;
        const float w11 = wx  * wy  * s;

        const int o00 = y0 * D_ + x0;
        const int o01 = y0 * D_ + x1;
        const int o10 = y1 * D_ + x0;
        const int o11 = y1 * D_ + x1;

#pragma unroll
        for (int c = 0; c < C_; ++c) {
            const float* __restrict__ sc = sino + (size_t)c * VD_;
            acc[c] = fmaf(sc[o00], w00, acc[c]);
            acc[c] = fmaf(sc[o01], w01, acc[c]);
            acc[c] = fmaf(sc[o10], w10, acc[c]);
            acc[c] = fmaf(sc[o11], w11, acc[c]);
        }
    }

    float* __restrict__ dst = dstbase + (size_t)seg * (C_ * P_);
#pragma unroll
    for (int c = 0; c < C_; ++c) dst[(size_t)c * P_ + p] = acc[c];
}

// Deterministic fixed-order reduction of view-segment partials.
__global__ __launch_bounds__(BLOCK) void IMGSR_reduce_kernel(
    const float* __restrict__ ws, float* __restrict__ out, int nseg)
{
    const int i = blockIdx.x * BLOCK + threadIdx.x;
    if (i < C_ * P_) {
        float a = 0.0f;
        for (int s = 0; s < nseg; ++s) a += ws[(size_t)s * (C_ * P_) + i];
        out[i] = a;
    }
}

extern "C" void kernel_launch(void* const* d_in, const int* in_sizes, int n_in,
                              void* d_out, int out_size, void* d_ws, size_t ws_size,
                              hipStream_t stream) {
    const float* sino = (const float*)d_in[0];
    const float* grid = (const float*)d_in[1];
    const float* sq   = (const float*)d_in[2];
    float* out        = (float*)d_out;

    const size_t need = (size_t)VSPLIT * C_ * P_ * sizeof(float);
    if (ws_size >= need) {
        dim3 g(P_ / BLOCK, VSPLIT);
        IMGSR_bp_kernel<V_ / VSPLIT><<<g, BLOCK, 0, stream>>>(sino, grid, sq, (float*)d_ws);
        IMGSR_reduce_kernel<<<(C_ * P_ + BLOCK - 1) / BLOCK, BLOCK, 0, stream>>>(
            (const float*)d_ws, out, VSPLIT);
    } else {
        dim3 g(P_ / BLOCK, 1);
        IMGSR_bp_kernel<V_><<<g, BLOCK, 0, stream>>>(sino, grid, sq, out);
    }
}